// PFGT_37555194036740
// MI455X (gfx1250) — compile-verified
//
#include <hip/hip_runtime.h>

#define NNODES 50000
#define NEDGES 400000
#define FIN 512
#define CDIM 64
#define CLSD 16
#define MSZ (CDIM * CLSD) // 1024 floats per node for M

typedef __attribute__((ext_vector_type(2))) float v2f;
typedef __attribute__((ext_vector_type(8))) float v8f;

__device__ __forceinline__ void atomic_add_f32(float* p, float v) {
    (void)__hip_atomic_fetch_add(p, v, __ATOMIC_RELAXED, __HIP_MEMORY_SCOPE_AGENT);
}

// ---------------------------------------------------------------------------
// h = relu(x @ W_in + b_in)   [N,512] x [512,64] -> [N,64]
// One wave computes a 16-row x 64-col strip with 4 WMMA accumulators.
// f32 WMMA 16x16x4 operand layout (wave32):
//   A 16x4 : lane l<16 holds row l, {K=k0,k0+1}; lane l>=16 holds row l-16, {K=k0+2,k0+3}
//   B 4x16 : lane l<16 holds col l, {K=k0,k0+1}; lane l>=16 holds col l-16, {K=k0+2,k0+3}
//   C 16x16: vgpr r, lane l<16 -> (M=r, N=l); lane l>=16 -> (M=r+8, N=l-16)
// ---------------------------------------------------------------------------
__global__ void __launch_bounds__(256) k_gemm_in(const float* __restrict__ x,
                                                 const float* __restrict__ W,
                                                 const float* __restrict__ bias,
                                                 float* __restrict__ h) {
    const int lane  = threadIdx.x & 31;
    const int strip = blockIdx.x * 8 + (threadIdx.x >> 5); // wave-uniform
    if (strip >= NNODES / 16) return;                      // EXEC stays all-ones
    const int half = lane >> 4;
    const int l    = lane & 15;
    const int row0 = strip * 16;
    const float* xrow = x + (size_t)(row0 + l) * FIN;

    v8f acc[4] = {};
    for (int k0 = 0; k0 < FIN; k0 += 4) {
        v2f a = *(const v2f*)(xrow + k0 + 2 * half);
        const float* wk = W + (size_t)(k0 + 2 * half) * CDIM + l;
#pragma unroll
        for (int t = 0; t < 4; ++t) {
            v2f b;
            b.x = wk[16 * t];
            b.y = wk[16 * t + CDIM];
            acc[t] = __builtin_amdgcn_wmma_f32_16x16x4_f32(
                false, a, false, b, (short)0, acc[t], false, false);
        }
    }
#pragma unroll
    for (int t = 0; t < 4; ++t) {
        const int col = 16 * t + l;
        const float bb = bias[col];
#pragma unroll
        for (int r = 0; r < 8; ++r) {
            float v = acc[t][r] + bb;
            h[(size_t)(row0 + r + 8 * half) * CDIM + col] = v > 0.f ? v : 0.f;
        }
    }
}

// ---------------------------------------------------------------------------
// Q = 1+elu(h@WQ+bQ), K = 1+elu(h@WK+bK), V = h@WV+bV   (K-dim = 64)
// ---------------------------------------------------------------------------
__global__ void __launch_bounds__(256) k_qkv(const float* __restrict__ h,
    const float* __restrict__ WQ, const float* __restrict__ bQ,
    const float* __restrict__ WK, const float* __restrict__ bK,
    const float* __restrict__ WV, const float* __restrict__ bV,
    float* __restrict__ Q, float* __restrict__ K, float* __restrict__ V) {
    const int lane  = threadIdx.x & 31;
    const int strip = blockIdx.x * 8 + (threadIdx.x >> 5);
    if (strip >= NNODES / 16) return;
    const int half = lane >> 4;
    const int l    = lane & 15;
    const int row0 = strip * 16;
    const float* hrow = h + (size_t)(row0 + l) * CDIM;

    v8f aq[4] = {}, ak[4] = {}, av = {};
    for (int k0 = 0; k0 < CDIM; k0 += 4) {
        v2f a = *(const v2f*)(hrow + k0 + 2 * half);
        const float* wq = WQ + (size_t)(k0 + 2 * half) * CDIM + l;
        const float* wk = WK + (size_t)(k0 + 2 * half) * CDIM + l;
        const float* wv = WV + (size_t)(k0 + 2 * half) * CLSD + l;
#pragma unroll
        for (int t = 0; t < 4; ++t) {
            v2f b;
            b.x = wq[16 * t];
            b.y = wq[16 * t + CDIM];
            aq[t] = __builtin_amdgcn_wmma_f32_16x16x4_f32(
                false, a, false, b, (short)0, aq[t], false, false);
            v2f c;
            c.x = wk[16 * t];
            c.y = wk[16 * t + CDIM];
            ak[t] = __builtin_amdgcn_wmma_f32_16x16x4_f32(
                false, a, false, c, (short)0, ak[t], false, false);
        }
        v2f bv2;
        bv2.x = wv[0];
        bv2.y = wv[CLSD];
        av = __builtin_amdgcn_wmma_f32_16x16x4_f32(
            false, a, false, bv2, (short)0, av, false, false);
    }
#pragma unroll
    for (int t = 0; t < 4; ++t) {
        const int col = 16 * t + l;
        const float bq = bQ[col];
        const float bk = bK[col];
#pragma unroll
        for (int r = 0; r < 8; ++r) {
            const size_t ro = (size_t)(row0 + r + 8 * half) * CDIM + col;
            float qv = aq[t][r] + bq;               // 1+elu(x) = 1+x (x>0) else exp(x)
            Q[ro] = qv > 0.f ? 1.f + qv : __expf(qv);
            float kv = ak[t][r] + bk;
            K[ro] = kv > 0.f ? 1.f + kv : __expf(kv);
        }
    }
    {
        const float bv = bV[l];
#pragma unroll
        for (int r = 0; r < 8; ++r)
            V[(size_t)(row0 + r + 8 * half) * CLSD + l] = av[r] + bv;
    }
}

// ---------------------------------------------------------------------------
// M0[n,i,:] = K[n,i] * V[n,:]  ;  hidden[n,:] = temp[0] * V[n,:]
// grid covers N*64 threads (one per (n,i))
// ---------------------------------------------------------------------------
__global__ void __launch_bounds__(256) k_init(const float* __restrict__ K,
                                              const float* __restrict__ V,
                                              const float* __restrict__ temp,
                                              float* __restrict__ M,
                                              float* __restrict__ out) {
    const int idx = blockIdx.x * 256 + threadIdx.x;
    const int n = idx >> 6, i = idx & 63;
    const float kv = K[(size_t)n * CDIM + i];
    const float* vrow = V + (size_t)n * CLSD;
    float* mrow = M + (size_t)n * MSZ + (size_t)i * CLSD;
#pragma unroll
    for (int j = 0; j < CLSD; ++j) mrow[j] = kv * vrow[j];
    if (i < CLSD) out[(size_t)n * CLSD + i] = temp[0] * vrow[i];
}

// ---------------------------------------------------------------------------
// Edge scatter-add: Mn[dst] += Mc[src] (1024 f32/edge), Kn[dst] += Kc[src].
// One thread per M element: wave handles 32 contiguous floats of one edge's
// tile -> uniform src/dst, coalesced loads, contiguous f32 atomics.
// ---------------------------------------------------------------------------
__global__ void __launch_bounds__(256) k_prop(const float* __restrict__ Mc,
                                              const float* __restrict__ Kc,
                                              float* __restrict__ Mn,
                                              float* __restrict__ Kn,
                                              const int* __restrict__ ei) {
    const int t = blockIdx.x * 256 + threadIdx.x;   // E*1024 total (<2^31)
    const int e = t >> 10, j = t & 1023;
    const int src = ei[e];
    const int dst = ei[NEDGES + e];
    atomic_add_f32(&Mn[(size_t)dst * MSZ + j], Mc[(size_t)src * MSZ + j]);
    if (j < CDIM)
        atomic_add_f32(&Kn[(size_t)dst * CDIM + j], Kc[(size_t)src * CDIM + j]);
}

// ---------------------------------------------------------------------------
// H[n,j] = sum_i Q[n,i]*M[n,i,j]; Cden = Q[n]·K[n] + 1e-5;
// hidden[n,j] += temp[hop+1] * H[n,j] / Cden.  16 lanes per node.
// ---------------------------------------------------------------------------
__global__ void __launch_bounds__(256) k_combine(const float* __restrict__ M,
                                                 const float* __restrict__ Q,
                                                 const float* __restrict__ K,
                                                 const float* __restrict__ temp,
                                                 int hop1,
                                                 float* __restrict__ out) {
    const int j = threadIdx.x & 15;
    const int n = blockIdx.x * 16 + (threadIdx.x >> 4); // N%16==0, exact
    const float* mrow = M + (size_t)n * MSZ;
    const float* qrow = Q + (size_t)n * CDIM;
    const float* krow = K + (size_t)n * CDIM;
    float hacc = 0.f, cacc = 0.f;
#pragma unroll 8
    for (int i = 0; i < CDIM; ++i) hacc += qrow[i] * mrow[i * CLSD + j];
#pragma unroll
    for (int tt = 0; tt < 4; ++tt) {
        const int i = j + 16 * tt;
        cacc += qrow[i] * krow[i];
    }
#pragma unroll
    for (int m = 1; m < 16; m <<= 1) cacc += __shfl_xor(cacc, m, 16);
    const float tv = temp[hop1];
    out[(size_t)n * CLSD + j] += tv * hacc / (cacc + 1e-5f);
}

// ---------------------------------------------------------------------------
extern "C" void kernel_launch(void* const* d_in, const int* in_sizes, int n_in,
                              void* d_out, int out_size, void* d_ws, size_t ws_size,
                              hipStream_t stream) {
    const float* x    = (const float*)d_in[0];
    const int*   ei   = (const int*)d_in[1];
    const float* W_in = (const float*)d_in[2];
    const float* b_in = (const float*)d_in[3];
    const float* WQ   = (const float*)d_in[4];
    const float* bQ   = (const float*)d_in[5];
    const float* WK   = (const float*)d_in[6];
    const float* bK   = (const float*)d_in[7];
    const float* WV   = (const float*)d_in[8];
    const float* bV   = (const float*)d_in[9];
    const float* temp = (const float*)d_in[10];
    float* out = (float*)d_out;

    // workspace layout (fp32): h | Q | K0 | K1 | V | M0 | M1  (~464 MB)
    float* ws = (float*)d_ws;
    const size_t N64 = (size_t)NNODES * CDIM;
    const size_t N16 = (size_t)NNODES * CLSD;
    const size_t NM  = (size_t)NNODES * MSZ;
    float* h  = ws;
    float* Q  = h + N64;
    float* K0 = Q + N64;
    float* K1 = K0 + N64;
    float* V  = K1 + N64;
    float* M0 = V + N16;
    float* M1 = M0 + NM;

    const int nstrips    = NNODES / 16;          // 3125
    const int gemmBlocks = (nstrips + 7) / 8;    // 391 blocks x 8 waves
    k_gemm_in<<<gemmBlocks, 256, 0, stream>>>(x, W_in, b_in, h);
    k_qkv<<<gemmBlocks, 256, 0, stream>>>(h, WQ, bQ, WK, bK, WV, bV, Q, K0, V);
    k_init<<<(int)(N64 / 256), 256, 0, stream>>>(K0, V, temp, M0, out);

    float* Mc = M0; float* Mn = M1;
    float* Kc = K0; float* Kn = K1;
    const int propBlocks = (int)((size_t)NEDGES * MSZ / 256); // 1,600,000
    for (int hop = 0; hop < 4; ++hop) {
        hipMemsetAsync(Mn, 0, NM * sizeof(float), stream);
        hipMemsetAsync(Kn, 0, N64 * sizeof(float), stream);
        k_prop<<<propBlocks, 256, 0, stream>>>(Mc, Kc, Mn, Kn, ei);
        k_combine<<<NNODES / 16, 256, 0, stream>>>(Mn, Q, Kn, temp, hop + 1, out);
        float* tm = Mc; Mc = Mn; Mn = tm;
        float* tk = Kc; Kc = Kn; Kn = tk;
    }
}